// QCAttention_73607149519303
// MI455X (gfx1250) — compile-verified
//
#include <hip/hip_runtime.h>
#include <math.h>

constexpr int NB = 16;    // batches
constexpr int LC = 1024;
constexpr int LQ = 128;
constexpr int DD = 512;
constexpr int NCH = 16;           // chunks over LC for column-softmax reduction
constexpr int CHROWS = LC / NCH;  // 64
#define NEGBIG (-1.0e10f)

typedef float v2f __attribute__((ext_vector_type(2)));
typedef float v8f __attribute__((ext_vector_type(8)));

#define WMMA_F32(a, b, c) \
  __builtin_amdgcn_wmma_f32_16x16x4_f32(false, (a), false, (b), (short)0, (c), false, false)

__device__ __forceinline__ float wave_sum32(float v) {
#pragma unroll
  for (int off = 16; off > 0; off >>= 1) v += __shfl_xor(v, off, 32);
  return v;
}
__device__ __forceinline__ float wave_max32(float v) {
#pragma unroll
  for (int off = 16; off > 0; off >>= 1) v = fmaxf(v, __shfl_xor(v, off, 32));
  return v;
}

// ---------------- Kernel 0: s_c = C@w1 (per row), s_q = Q@w2 ----------------
// one wave per row; grid = (NB*LC + NB*LQ)/8 blocks of 256
__global__ void k_rowscores(const float* __restrict__ C, const float* __restrict__ Q,
                            const float* __restrict__ w, float* __restrict__ sc,
                            float* __restrict__ sq) {
  const int wv = threadIdx.x >> 5, lane = threadIdx.x & 31;
  const int r = blockIdx.x * 8 + wv;
  const float* x;
  const float* wp;
  float* dst;
  if (r < NB * LC) {
    x = C + (size_t)r * DD; wp = w; dst = sc + r;
  } else {
    const int r2 = r - NB * LC;
    x = Q + (size_t)r2 * DD; wp = w + DD; dst = sq + r2;
  }
  float s = 0.f;
#pragma unroll 4
  for (int d = lane; d < DD; d += 32) s += x[d] * wp[d];
  s = wave_sum32(s);
  if (lane == 0) *dst = s;
}

// ---------------- Kernel 1: S = (C*w3) @ Q^T + sc + sq ----------------
// grid (LC/64, NB), 256 threads = 8 waves; wave (wm 0..3, wn 0..1) -> 16x64 tile
__global__ void k_scores(const float* __restrict__ C, const float* __restrict__ Q,
                         const float* __restrict__ w, const float* __restrict__ sc,
                         const float* __restrict__ sq, float* __restrict__ S) {
  const int b = blockIdx.y;
  const int wv = threadIdx.x >> 5, lane = threadIdx.x & 31;
  const int wm = wv & 3, wn = wv >> 2;
  const int i0 = blockIdx.x * 64 + wm * 16;
  const int j0 = wn * 64;
  const int lh = lane >> 4, lm = lane & 15;

  const float* Crow = C + ((size_t)b * LC + (i0 + lm)) * DD + 2 * lh;
  const float* w3 = w + 2 * DD + 2 * lh;
  const float* Qrow = Q + ((size_t)b * LQ + (j0 + lm)) * DD + 2 * lh;

  v8f acc[4] = {v8f{}, v8f{}, v8f{}, v8f{}};
  for (int k = 0; k < DD; k += 4) {
    v2f a;
    a.x = Crow[k] * w3[k];
    a.y = Crow[k + 1] * w3[k + 1];
#pragma unroll
    for (int nt = 0; nt < 4; ++nt) {
      v2f bb;
      bb.x = Qrow[(size_t)nt * 16 * DD + k];
      bb.y = Qrow[(size_t)nt * 16 * DD + k + 1];
      acc[nt] = WMMA_F32(a, bb, acc[nt]);
    }
  }
#pragma unroll
  for (int v = 0; v < 8; ++v) {
    const int i = i0 + v + 8 * lh;
    const float sci = sc[b * LC + i];
#pragma unroll
    for (int nt = 0; nt < 4; ++nt) {
      const int j = j0 + nt * 16 + lm;
      S[((size_t)b * LC + i) * LQ + j] = acc[nt][v] + sci + sq[b * LQ + j];
    }
  }
}

// ---------------- Kernel 2a: per-chunk column stats (online max/sum over 64 rows) ----------
// grid (NCH, NB), 128 threads; thread owns column j within chunk ch
__global__ void k_colstats1(const float* __restrict__ S, const int* __restrict__ Cmask,
                            float* __restrict__ pmax, float* __restrict__ psum) {
  const int b = blockIdx.y;
  const int ch = blockIdx.x;
  const int j = threadIdx.x;
  const float* p = S + (size_t)b * LC * LQ + (size_t)ch * CHROWS * LQ + j;
  const int* mb = Cmask + b * LC + ch * CHROWS;
  float m = -INFINITY, s = 0.f;
  for (int i = 0; i < CHROWS; ++i) {
    const float v = (mb[i] != 0) ? NEGBIG : p[(size_t)i * LQ];
    const float nm = fmaxf(m, v);
    s = s * __expf(m - nm) + __expf(v - nm);
    m = nm;
  }
  pmax[(b * NCH + ch) * LQ + j] = m;
  psum[(b * NCH + ch) * LQ + j] = s;
}

// ---------------- Kernel 2b: combine chunk stats -> cmax, cinv ----------------
// grid NB, 128 threads
__global__ void k_colstats2(const float* __restrict__ pmax, const float* __restrict__ psum,
                            float* __restrict__ cmax, float* __restrict__ cinv) {
  const int b = blockIdx.x;
  const int j = threadIdx.x;
  float m = -INFINITY;
#pragma unroll
  for (int ch = 0; ch < NCH; ++ch) m = fmaxf(m, pmax[(b * NCH + ch) * LQ + j]);
  float s = 0.f;
#pragma unroll
  for (int ch = 0; ch < NCH; ++ch)
    s += psum[(b * NCH + ch) * LQ + j] * __expf(pmax[(b * NCH + ch) * LQ + j] - m);
  cmax[b * LQ + j] = m;
  cinv[b * LQ + j] = 1.f / s;
}

// ---------------- Kernel 2c: S2 = exp(masked S - cmax) * cinv (elementwise) ----------------
// grid NB*LC*LQ/1024, 256 threads, 4 contiguous elems per thread
__global__ void k_writes2(const float* __restrict__ S, const int* __restrict__ Cmask,
                          const float* __restrict__ cmax, const float* __restrict__ cinv,
                          float* __restrict__ S2) {
  const size_t e0 = ((size_t)blockIdx.x * blockDim.x + threadIdx.x) * 4;
#pragma unroll
  for (int t = 0; t < 4; ++t) {
    const size_t e = e0 + t;
    const int j = (int)(e & (LQ - 1));
    const size_t bi = e >> 7;           // b*LC + i
    const int b = (int)(bi >> 10);      // / LC
    const float v = (Cmask[bi] != 0) ? NEGBIG : S[e];
    S2[e] = __expf(v - cmax[b * LQ + j]) * cinv[b * LQ + j];
  }
}

// ---------------- Kernel 3: row softmax over LQ (Qmask), in place -> S1 ----------------
// one wave per (b,i) row; grid NB*LC/8 blocks of 256
__global__ void k_rowsoftmax(float* __restrict__ S, const int* __restrict__ Qmask) {
  const int wv = threadIdx.x >> 5, lane = threadIdx.x & 31;
  const size_t row = (size_t)blockIdx.x * 8 + wv;  // over NB*LC
  const int b = (int)(row / LC);
  float* Sr = S + row * LQ;
  const int* qm = Qmask + b * LQ;
  float v[4];
  float m = -INFINITY;
#pragma unroll
  for (int t = 0; t < 4; ++t) {
    const int j = lane + 32 * t;
    v[t] = (qm[j] != 0) ? NEGBIG : Sr[j];
    m = fmaxf(m, v[t]);
  }
  m = wave_max32(m);
  float s = 0.f;
#pragma unroll
  for (int t = 0; t < 4; ++t) {
    v[t] = __expf(v[t] - m);
    s += v[t];
  }
  s = wave_sum32(s);
  const float inv = 1.f / s;
#pragma unroll
  for (int t = 0; t < 4; ++t) Sr[lane + 32 * t] = v[t] * inv;
}

// ---------------- Kernel 4: T = S2^T @ C  (M=LQ, N=DD, K=LC) ----------------
// grid (8, NB): tile t -> tileM = t&3 (32 rows), tileN = t>>2 (256 cols); 8 waves
__global__ void k_s2tc(const float* __restrict__ S2, const float* __restrict__ C,
                       float* __restrict__ T) {
  const int b = blockIdx.y;
  const int t = blockIdx.x;
  const int tileM = t & 3, tileN = t >> 2;
  const int wv = threadIdx.x >> 5, lane = threadIdx.x & 31;
  const int wm = wv & 1, wn = wv >> 1;
  const int j0 = tileM * 32 + wm * 16;
  const int d0 = tileN * 256 + wn * 64;
  const int lh = lane >> 4, lm = lane & 15;
  const float* S2b = S2 + (size_t)b * LC * LQ;
  const float* Cb = C + (size_t)b * LC * DD;

  v8f acc[4] = {v8f{}, v8f{}, v8f{}, v8f{}};
  for (int k = 0; k < LC; k += 4) {
    const int kr = k + 2 * lh;
    v2f a;
    a.x = S2b[(size_t)kr * LQ + j0 + lm];
    a.y = S2b[(size_t)(kr + 1) * LQ + j0 + lm];
#pragma unroll
    for (int nt = 0; nt < 4; ++nt) {
      const int d = d0 + nt * 16 + lm;
      v2f bb;
      bb.x = Cb[(size_t)kr * DD + d];
      bb.y = Cb[(size_t)(kr + 1) * DD + d];
      acc[nt] = WMMA_F32(a, bb, acc[nt]);
    }
  }
#pragma unroll
  for (int v = 0; v < 8; ++v) {
    const int j = j0 + v + 8 * lh;
#pragma unroll
    for (int nt = 0; nt < 4; ++nt) {
      const int d = d0 + nt * 16 + lm;
      T[((size_t)b * LQ + j) * DD + d] = acc[nt][v];
    }
  }
}

// ---------------- Kernel 5: A = S1@Q, Bv = S1@T, out = [C, A, C*A, C*Bv] ----------------
// grid (LC/16, NB), 256 threads; wave wv owns 64 output cols; fused dual GEMM, K=LQ
__global__ void k_final(const float* __restrict__ S1, const float* __restrict__ Q,
                        const float* __restrict__ T, const float* __restrict__ C,
                        float* __restrict__ out) {
  const int b = blockIdx.y;
  const int i0 = blockIdx.x * 16;
  const int wv = threadIdx.x >> 5, lane = threadIdx.x & 31;
  const int d0 = wv * 64;
  const int lh = lane >> 4, lm = lane & 15;

  const float* S1row = S1 + ((size_t)b * LC + (i0 + lm)) * LQ + 2 * lh;
  const float* Qb = Q + (size_t)b * LQ * DD;
  const float* Tb = T + (size_t)b * LQ * DD;

  v8f accA[4] = {v8f{}, v8f{}, v8f{}, v8f{}};
  v8f accB[4] = {v8f{}, v8f{}, v8f{}, v8f{}};
  for (int k = 0; k < LQ; k += 4) {
    const v2f a = *(const v2f*)(S1row + k);  // contiguous K: 8B aligned
    const int kr = k + 2 * lh;
#pragma unroll
    for (int nt = 0; nt < 4; ++nt) {
      const int d = d0 + nt * 16 + lm;
      v2f bq, bt;
      bq.x = Qb[(size_t)kr * DD + d];
      bq.y = Qb[(size_t)(kr + 1) * DD + d];
      bt.x = Tb[(size_t)kr * DD + d];
      bt.y = Tb[(size_t)(kr + 1) * DD + d];
      accA[nt] = WMMA_F32(a, bq, accA[nt]);
      accB[nt] = WMMA_F32(a, bt, accB[nt]);
    }
  }
  const float* Cb = C + (size_t)b * LC * DD;
  float* ob = out + (size_t)b * LC * (4 * DD);
#pragma unroll
  for (int v = 0; v < 8; ++v) {
    const int i = i0 + v + 8 * lh;
    float* orow = ob + (size_t)i * (4 * DD);
#pragma unroll
    for (int nt = 0; nt < 4; ++nt) {
      const int d = d0 + nt * 16 + lm;
      const float c = Cb[(size_t)i * DD + d];
      const float aV = accA[nt][v];
      const float bV = accB[nt][v];
      orow[d] = c;
      orow[DD + d] = aV;
      orow[2 * DD + d] = c * aV;
      orow[3 * DD + d] = c * bV;
    }
  }
}

extern "C" void kernel_launch(void* const* d_in, const int* in_sizes, int n_in,
                              void* d_out, int out_size, void* d_ws, size_t ws_size,
                              hipStream_t stream) {
  (void)in_sizes; (void)n_in; (void)out_size; (void)ws_size;
  const float* C = (const float*)d_in[0];
  const float* Q = (const float*)d_in[1];
  const int* Cmask = (const int*)d_in[2];
  const int* Qmask = (const int*)d_in[3];
  const float* w = (const float*)d_in[4];
  float* out = (float*)d_out;

  // workspace layout (floats): S | S2 | T | sc | sq | pmax | psum | cmax | cinv  (~21.3 MB)
  float* ws = (float*)d_ws;
  float* S    = ws;
  float* S2   = S + (size_t)NB * LC * LQ;
  float* T    = S2 + (size_t)NB * LC * LQ;
  float* sc   = T + (size_t)NB * LQ * DD;
  float* sq   = sc + (size_t)NB * LC;
  float* pmax = sq + (size_t)NB * LQ;
  float* psum = pmax + (size_t)NB * NCH * LQ;
  float* cmax = psum + (size_t)NB * NCH * LQ;
  float* cinv = cmax + (size_t)NB * LQ;

  k_rowscores<<<(NB * LC + NB * LQ) / 8, 256, 0, stream>>>(C, Q, w, sc, sq);
  k_scores<<<dim3(LC / 64, NB), 256, 0, stream>>>(C, Q, w, sc, sq, S);
  k_colstats1<<<dim3(NCH, NB), LQ, 0, stream>>>(S, Cmask, pmax, psum);
  k_colstats2<<<NB, LQ, 0, stream>>>(pmax, psum, cmax, cinv);
  k_writes2<<<(NB * LC * LQ) / 1024, 256, 0, stream>>>(S, Cmask, cmax, cinv, S2);
  k_rowsoftmax<<<NB * LC / 8, 256, 0, stream>>>(S, Qmask);
  k_s2tc<<<dim3(8, NB), 256, 0, stream>>>(S2, C, T);
  k_final<<<dim3(LC / 16, NB), 256, 0, stream>>>(S, Q, T, C, out);
}